// RotatedMultiboxLoss_17592186045046
// MI455X (gfx1250) — compile-verified
//
#include <hip/hip_runtime.h>
#include <cstdint>
#include <cstddef>

// Problem constants (match reference).
#define B_     32
#define N_     32768
#define C_     21
#define RATIO_ 3

// Accumulator block lives in d_ws right after the key array.
struct Accum {
    int   posb[B_];    // positives per batch row
    int   pos_total;   // total positives
    float sl1;         // summed SmoothL1 over positives
    float ce_pos;      // summed CE over positives
    float ce_neg;      // summed CE over mined negatives (top-k mining losses)
};

#define KEY_BYTES ((size_t)B_ * N_ * sizeof(uint32_t))

// ---------------------------------------------------------------------------
// Kernel 0: zero the accumulators (replay-safe; keys are fully overwritten).
// ---------------------------------------------------------------------------
__global__ void init_ws(Accum* __restrict__ acc) {
    int t = threadIdx.x;
    if (t < B_) acc->posb[t] = 0;
    if (t == 0) {
        acc->pos_total = 0;
        acc->sl1 = 0.0f;
        acc->ce_pos = 0.0f;
        acc->ce_neg = 0.0f;
    }
}

// ---------------------------------------------------------------------------
// Kernel A: per-prior logsumexp, keys for mining, positive-side reductions.
// One thread per (b, n); 256 threads/block = 8 wave32 waves. The block's
// 256x21-float confidence tile (21 KB) is staged global->LDS with coalesced
// CDNA5 async-to-LDS copies (128 B per wave instruction), then each thread
// consumes its own row from LDS (stride 21 dwords: odd, so conflict-free
// across the 64 LDS banks). N % 256 == 0, so each block lies inside one
// batch row and positive counts map to a single posb[b] atomic.
// ---------------------------------------------------------------------------
__global__ __launch_bounds__(256) void pass_logsumexp(
    const float* __restrict__ conf,   // [B*N, C]
    const float* __restrict__ ploc,   // [B*N, 5]
    const float* __restrict__ gloc,   // [B*N, 5]
    const int*   __restrict__ labels, // [B*N]
    uint32_t*    __restrict__ keys,   // [B*N] monotone-uint mining keys
    Accum*       __restrict__ acc) {

    __shared__ float ctile[256 * C_];                   // 21504 B staging tile

    const int tid = threadIdx.x;
    const int i   = blockIdx.x * 256 + tid;             // 0 .. B*N-1

    // --- stage the block's confidence tile, coalesced, via async DMA ---
    const uint64_t gbase =
        (uint64_t)(uintptr_t)(conf + (size_t)blockIdx.x * 256 * C_);
#pragma unroll
    for (int j = 0; j < C_; ++j) {
        const uint32_t lin  = (uint32_t)(j * 256 + tid); // linear dword in tile
        const uint32_t voff = lin * 4u;                  // global byte offset
        const uint32_t ldsa = (uint32_t)(uintptr_t)&ctile[lin];
        asm volatile("global_load_async_to_lds_b32 %0, %1, %2 offset:0"
                     :: "v"(ldsa), "v"(voff), "s"(gbase) : "memory");
    }
    asm volatile("s_wait_asynccnt 0" ::: "memory");      // this wave's copies done
    __syncthreads();                                     // => all waves' copies done

    // --- streaming logsumexp over this thread's 21 logits (from LDS) ---
    float x[C_];
#pragma unroll
    for (int c = 0; c < C_; ++c) x[c] = ctile[tid * C_ + c];

    float m = x[0];
#pragma unroll
    for (int c = 1; c < C_; ++c) m = fmaxf(m, x[c]);
    float s = 0.0f;
#pragma unroll
    for (int c = 0; c < C_; ++c) s += __expf(x[c] - m);
    const float lse = m + __logf(s);                    // logsumexp

    const int lab = labels[i];
    const float mining = lse - x[0];                    // -log softmax[0] >= 0

    float ce = 0.0f, sl1 = 0.0f;
    int isp = 0;
    uint32_t key;
    if (lab > 0) {
        isp = 1;
        ce = lse - ctile[tid * C_ + lab];               // dynamic gather via LDS
        key = 0x007FFFFFu;                              // order(-inf): excluded
        const float* p = ploc + (size_t)i * 5;
        const float* g = gloc + (size_t)i * 5;
#pragma unroll
        for (int j = 0; j < 5; ++j) {
            float d = fabsf(p[j] - g[j]);
            sl1 += (d < 1.0f) ? 0.5f * d * d : d - 0.5f;
        }
    } else {
        // Monotone float->uint map (bigger float => bigger uint).
        uint32_t bits = __float_as_uint(mining);
        key = (bits & 0x80000000u) ? ~bits : (bits | 0x80000000u);
    }
    keys[i] = key;

    // wave32 shuffle reduction, then 8-wave LDS combine, then 1 atomic each.
    float v0 = sl1, v1 = ce;
    int   v2 = isp;
    for (int o = 16; o > 0; o >>= 1) {
        v0 += __shfl_xor(v0, o);
        v1 += __shfl_xor(v1, o);
        v2 += __shfl_xor(v2, o);
    }
    __shared__ float s0[8], s1[8];
    __shared__ int   s2[8];
    const int wid = tid >> 5, lane = tid & 31;
    if (lane == 0) { s0[wid] = v0; s1[wid] = v1; s2[wid] = v2; }
    __syncthreads();
    if (tid == 0) {
        float a0 = 0.0f, a1 = 0.0f; int a2 = 0;
#pragma unroll
        for (int w = 0; w < 8; ++w) { a0 += s0[w]; a1 += s1[w]; a2 += s2[w]; }
        if (a0 != 0.0f) atomicAdd(&acc->sl1, a0);
        if (a1 != 0.0f) atomicAdd(&acc->ce_pos, a1);
        if (a2 != 0) {
            atomicAdd(&acc->pos_total, a2);
            atomicAdd(&acc->posb[i / N_], a2);          // block is inside one b
        }
    }
}

// ---------------------------------------------------------------------------
// Kernel B: per batch row, 4-pass MSB-first radix select of the top-k mining
// losses among negatives; accumulates their float sum (== mined-negative CE,
// since CE == mining loss for background priors). Key tiles are staged
// global->LDS with the CDNA5 async-to-LDS path.
// ---------------------------------------------------------------------------
__global__ __launch_bounds__(1024) void select_topk(
    const uint32_t* __restrict__ keys, Accum* __restrict__ acc) {

    const int b   = blockIdx.x;
    const int tid = threadIdx.x;

    __shared__ unsigned int hist[256];   // per-digit counts
    __shared__ float        hsum[256];   // per-digit float value sums (ds_add_f32)
    __shared__ uint32_t     tile[1024];  // async-staged key tile (private slot per thread)
    __shared__ uint32_t     sh_digit;
    __shared__ int          sh_k;
    __shared__ float        sh_add;

    const int npos = acc->posb[b];
    int k = npos * RATIO_;
    const int nneg = N_ - npos;
    if (k > nneg) k = nneg;
    if (k <= 0) return;                  // uniform across block: safe early-out

    const uint64_t gbase = (uint64_t)(uintptr_t)(keys + (size_t)b * N_);

    uint32_t prefix = 0;                 // resolved high bits of the threshold key
    float included = 0.0f;               // sum of values definitely in top-k
    int kr = k;                          // still-needed count

    for (int pass = 0; pass < 4; ++pass) {
        const int shift = 24 - 8 * pass;
        const uint32_t himask =
            (uint32_t)((pass == 0) ? 0ull : (0xFFFFFFFFull << (shift + 8)));

        if (tid < 256) { hist[tid] = 0u; hsum[tid] = 0.0f; }
        __syncthreads();

        for (int base0 = 0; base0 < N_; base0 += 1024) {
            // CDNA5 async global->LDS copy of this thread's key.
            const uint32_t voff = (uint32_t)((base0 + tid) * 4);
            const uint32_t ldsa = (uint32_t)(uintptr_t)&tile[tid];
            asm volatile("global_load_async_to_lds_b32 %0, %1, %2 offset:0"
                         :: "v"(ldsa), "v"(voff), "s"(gbase) : "memory");
            asm volatile("s_wait_asynccnt 0" ::: "memory");
            const uint32_t u = tile[tid];   // own slot only: no barrier needed

            if (((u ^ prefix) & himask) == 0u) {
                const uint32_t d = (u >> shift) & 255u;
                atomicAdd(&hist[d], 1u);
                const uint32_t fb =
                    (u & 0x80000000u) ? (u ^ 0x80000000u) : ~u;  // inverse map
                atomicAdd(&hsum[d], __uint_as_float(fb));        // ds_add_f32
            }
        }
        __syncthreads();

        if (tid == 0) {
            int cum = 0; float addv = 0.0f; int d;
            for (d = 255; d >= 0; --d) {
                const int c = (int)hist[d];
                if (cum + c >= kr) break;      // kr-th largest lands in bin d
                cum += c; addv += hsum[d];     // bins above d: all included
            }
            sh_digit = (uint32_t)d;
            sh_k     = kr - cum;
            sh_add   = addv;
        }
        __syncthreads();

        included += sh_add;
        kr = sh_k;
        prefix |= (sh_digit << shift);
    }

    if (tid == 0) {
        // kr elements are exactly equal to the threshold key -> equal values.
        const uint32_t u = prefix;
        const uint32_t fb = (u & 0x80000000u) ? (u ^ 0x80000000u) : ~u;
        included += (float)kr * __uint_as_float(fb);
        atomicAdd(&acc->ce_neg, included);
    }
}

// ---------------------------------------------------------------------------
// Kernel C: finalize the two output scalars.
// ---------------------------------------------------------------------------
__global__ void finalize(const Accum* __restrict__ acc, float* __restrict__ out) {
    const float np = (float)acc->pos_total;
    out[0] = acc->sl1 / np;
    out[1] = (acc->ce_pos + acc->ce_neg) / np;
}

// ---------------------------------------------------------------------------
// Launch: inputs are (confidence, predicted_locations, labels, gt_locations).
// d_ws usage: [0, 4MB) key array, then Accum (~160B).
// ---------------------------------------------------------------------------
extern "C" void kernel_launch(void* const* d_in, const int* in_sizes, int n_in,
                              void* d_out, int out_size, void* d_ws, size_t ws_size,
                              hipStream_t stream) {
    const float* conf   = (const float*)d_in[0];
    const float* ploc   = (const float*)d_in[1];
    const int*   labels = (const int*)d_in[2];
    const float* gloc   = (const float*)d_in[3];

    uint32_t* keys = (uint32_t*)d_ws;
    Accum*    acc  = (Accum*)((char*)d_ws + KEY_BYTES);

    init_ws<<<1, 64, 0, stream>>>(acc);
    pass_logsumexp<<<(B_ * N_) / 256, 256, 0, stream>>>(conf, ploc, gloc, labels,
                                                        keys, acc);
    select_topk<<<B_, 1024, 0, stream>>>(keys, acc);
    finalize<<<1, 1, 0, stream>>>(acc, (float*)d_out);
}